// EquivariantProductBasisBlock_42210938585330
// MI455X (gfx1250) — compile-verified
//
#include <hip/hip_runtime.h>
#include <hip/hip_bf16.h>

// ---------------- problem constants (match reference) ----------------
constexpr int N_   = 2048;   // nodes
constexpr int C_   = 128;    // channels
constexpr int I_   = 9;      // num_ell
constexpr int E_   = 10;     // elements
constexpr int K3_  = 23, K2_ = 8, K1_ = 3;
constexpr int DTOT = 4;      // d=0 -> irrep 0e (D0=1), d=1..3 -> irrep 1o (D1=3)
constexpr int OUTC = 512;    // C*(D0+D1)

// workspace layout (floats)
constexpr size_t A3_ELEMS = (size_t)E_ * DTOT * C_ * 729;   // 3,732,480
constexpr size_t A2_ELEMS = (size_t)E_ * DTOT * C_ * 81;    //   414,720
constexpr size_t A1_ELEMS = (size_t)E_ * DTOT * C_ * 9;     //    46,080
constexpr size_t FB_ELEMS = (size_t)DTOT * N_ * C_;         // 1,048,576

// ---------------- WMMA vector types ----------------
typedef __attribute__((ext_vector_type(2))) float v2f;
typedef __attribute__((ext_vector_type(8))) float v8f;

// ---------------- async global->LDS path (gfx1250) ----------------
#if __has_builtin(__builtin_amdgcn_global_load_async_to_lds_b128) && \
    __has_builtin(__builtin_amdgcn_s_wait_asynccnt)
#define USE_ASYNC_LDS 1
typedef int v4i __attribute__((ext_vector_type(4)));
typedef __attribute__((address_space(1))) v4i gv4i;   // global 16B element
typedef __attribute__((address_space(3))) v4i lv4i;   // LDS    16B element
#else
#define USE_ASYNC_LDS 0
#endif

// =====================================================================
// Stage 0: zero the per-element bucket counters
// =====================================================================
__global__ void zero_counts(int* counts) {
    if (threadIdx.x < 16) counts[threadIdx.x] = 0;
}

// =====================================================================
// Stage 1: bucket nodes by element type (node_attrs is one-hot [N,E])
// =====================================================================
__global__ void bucket_nodes(const float* __restrict__ attrs,
                             int* __restrict__ counts, int* __restrict__ lists) {
    int n = blockIdx.x * 256 + threadIdx.x;
    if (n >= N_) return;
    int e = 0;
    #pragma unroll
    for (int j = 0; j < E_; ++j)
        if (attrs[n * E_ + j] > 0.5f) e = j;
    int slot = atomicAdd(&counts[e], 1);
    lists[e * N_ + slot] = n;
}

// =====================================================================
// Stage 2: fold element weights into the CG tensors. (e,d,c,r) layouts
// make each block's staging chunk contiguous + 16B aligned:
//   A3[((e*4+d)*C + c)*729 + p*81+q*9+i] = sum_k U3[d,p,q,i,k]*W3[e,k,c]
//   A2[((e*4+d)*C + c)*81  + p*9+q]      = sum_k U2[d,p,q,k]  *W2[e,k,c]
//   A1[((e*4+d)*C + c)*9   + p]          = sum_k U1[d,p,k]    *W1[e,k,c]
// =====================================================================
__global__ void build_A3(const float* __restrict__ U3_0, const float* __restrict__ W3_0,
                         const float* __restrict__ U3_1, const float* __restrict__ W3_1,
                         float* __restrict__ A3) {
    size_t idx = (size_t)blockIdx.x * 256 + threadIdx.x;
    if (idx >= A3_ELEMS) return;
    size_t t = idx;
    int r = t % 729; t /= 729;          // r = p*81 + q*9 + i
    int c = t % C_;  t /= C_;
    int d = t % 4;
    int e = (int)(t / 4);
    float s = 0.f;
    if (d == 0) {
        const float* u = U3_0 + (size_t)r * K3_;
        const float* w = W3_0 + (size_t)e * K3_ * C_ + c;
        #pragma unroll
        for (int k = 0; k < K3_; ++k) s = fmaf(u[k], w[(size_t)k * C_], s);
    } else {
        const float* u = U3_1 + ((size_t)(d - 1) * 729 + r) * K3_;
        const float* w = W3_1 + (size_t)e * K3_ * C_ + c;
        #pragma unroll
        for (int k = 0; k < K3_; ++k) s = fmaf(u[k], w[(size_t)k * C_], s);
    }
    A3[idx] = s;
}

__global__ void build_A2(const float* __restrict__ U2_0, const float* __restrict__ W2_0,
                         const float* __restrict__ U2_1, const float* __restrict__ W2_1,
                         float* __restrict__ A2) {
    size_t idx = (size_t)blockIdx.x * 256 + threadIdx.x;
    if (idx >= A2_ELEMS) return;
    size_t t = idx;
    int r = t % 81; t /= 81;            // r = p*9 + q
    int c = t % C_; t /= C_;
    int d = t % 4;
    int e = (int)(t / 4);
    float s = 0.f;
    if (d == 0) {
        const float* u = U2_0 + (size_t)r * K2_;
        const float* w = W2_0 + (size_t)e * K2_ * C_ + c;
        #pragma unroll
        for (int k = 0; k < K2_; ++k) s = fmaf(u[k], w[(size_t)k * C_], s);
    } else {
        const float* u = U2_1 + ((size_t)(d - 1) * 81 + r) * K2_;
        const float* w = W2_1 + (size_t)e * K2_ * C_ + c;
        #pragma unroll
        for (int k = 0; k < K2_; ++k) s = fmaf(u[k], w[(size_t)k * C_], s);
    }
    A2[idx] = s;
}

__global__ void build_A1(const float* __restrict__ U1_0, const float* __restrict__ W1_0,
                         const float* __restrict__ U1_1, const float* __restrict__ W1_1,
                         float* __restrict__ A1) {
    size_t idx = (size_t)blockIdx.x * 256 + threadIdx.x;
    if (idx >= A1_ELEMS) return;
    size_t t = idx;
    int r = t % 9;  t /= 9;             // r = p
    int c = t % C_; t /= C_;
    int d = t % 4;
    int e = (int)(t / 4);
    float s = 0.f;
    if (d == 0) {
        const float* u = U1_0 + (size_t)r * K1_;
        const float* w = W1_0 + (size_t)e * K1_ * C_ + c;
        #pragma unroll
        for (int k = 0; k < K1_; ++k) s = fmaf(u[k], w[(size_t)k * C_], s);
    } else {
        const float* u = U1_1 + ((size_t)(d - 1) * 9 + r) * K1_;
        const float* w = W1_1 + (size_t)e * K1_ * C_ + c;
        #pragma unroll
        for (int k = 0; k < K1_; ++k) s = fmaf(u[k], w[(size_t)k * C_], s);
    }
    A1[idx] = s;
}

// =====================================================================
// Stage 3: symmetric contraction.
// Block = (element e, 16-channel tile, node slice). Per d, the A3/A2/A1
// slices (46.6KB + 5.2KB + 0.6KB) are one contiguous, 16B-aligned chunk
// -> staged with GLOBAL_LOAD_ASYNC_TO_LDS_B128 (ASYNCcnt), no VGPR
// round-trip. 16 nodes x 16 channels per step; threads sharing c_local
// hit the same LDS word -> broadcast.
//   f[d,n,c] = sum_p x_p*(A1 + sum_q x_q*(A2 + sum_i x_i*A3))
// =====================================================================
__global__ __launch_bounds__(256)
void sym_contract(const float* __restrict__ x,          // [N,C,9]
                  const int* __restrict__ counts, const int* __restrict__ lists,
                  const float* __restrict__ A3, const float* __restrict__ A2,
                  const float* __restrict__ A1, float* __restrict__ FB) {
    __shared__ __align__(16) float sA3[16 * 729];
    __shared__ __align__(16) float sA2[16 * 81];
    __shared__ __align__(16) float sA1[16 * 9];

    int b     = blockIdx.x;
    int slice = b & 15;          // 16 node-slices per (e, ctile)
    int ctile = (b >> 4) & 7;    // 8 channel tiles of 16
    int e     = b >> 7;          // 10 elements
    int cbase = ctile * 16;
    int tid   = threadIdx.x;
    int n_local = tid >> 4;
    int c_local = tid & 15;

    int nn  = counts[e];
    int per = (nn + 15) >> 4;
    int s0  = slice * per;
    int s1  = s0 + per; if (s1 > nn) s1 = nn;

    for (int d = 0; d < DTOT; ++d) {
        if (d) __syncthreads();
        const float* g3 = A3 + ((size_t)(e * 4 + d) * C_ + cbase) * 729;
        const float* g2 = A2 + ((size_t)(e * 4 + d) * C_ + cbase) * 81;
        const float* g1 = A1 + ((size_t)(e * 4 + d) * C_ + cbase) * 9;
#if USE_ASYNC_LDS
        {
            // per-lane 16-byte transfers: 2916 + 324 + 36 per d, all aligned
            gv4i* gg3 = (gv4i*)g3;
            gv4i* gg2 = (gv4i*)g2;
            gv4i* gg1 = (gv4i*)g1;
            lv4i* ll3 = (lv4i*)sA3;
            lv4i* ll2 = (lv4i*)sA2;
            lv4i* ll1 = (lv4i*)sA1;
            for (int l = tid; l < 2916; l += 256)
                __builtin_amdgcn_global_load_async_to_lds_b128(gg3 + l, ll3 + l, 0, 0);
            for (int l = tid; l < 324; l += 256)
                __builtin_amdgcn_global_load_async_to_lds_b128(gg2 + l, ll2 + l, 0, 0);
            if (tid < 36)
                __builtin_amdgcn_global_load_async_to_lds_b128(gg1 + tid, ll1 + tid, 0, 0);
            __builtin_amdgcn_s_wait_asynccnt(0);
        }
#else
        for (int l = tid; l < 16 * 729; l += 256) sA3[l] = g3[l];
        for (int l = tid; l < 16 * 81;  l += 256) sA2[l] = g2[l];
        if (tid < 16 * 9) sA1[tid] = g1[tid];
#endif
        __syncthreads();

        for (int t0 = s0; t0 < s1; t0 += 16) {
            int ni = t0 + n_local;
            if (ni < s1) {
                int n = lists[e * N_ + ni];
                const float* xp = x + ((size_t)n * C_ + cbase + c_local) * I_;
                float xv[9];
                #pragma unroll
                for (int i = 0; i < 9; ++i) xv[i] = xp[i];

                const float* a3 = sA3 + c_local * 729;
                const float* a2 = sA2 + c_local * 81;
                const float* a1 = sA1 + c_local * 9;
                float acc = 0.f;
                for (int p = 0; p < 9; ++p) {
                    float tp = a1[p];
                    for (int q = 0; q < 9; ++q) {
                        float tq = a2[p * 9 + q];
                        #pragma unroll
                        for (int i = 0; i < 9; ++i)
                            tq = fmaf(a3[p * 81 + q * 9 + i], xv[i], tq);
                        tp = fmaf(tq, xv[q], tp);
                    }
                    acc = fmaf(tp, xv[p], acc);
                }
                FB[((size_t)d * N_ + n) * C_ + cbase + c_local] = acc;
            }
        }
    }
}

// =====================================================================
// Stage 4: equivariant Linear via V_WMMA_F32_16X16X4_F32 (wave32).
// One wave computes one 16x16 tile of Y_d = F_d @ Wlin * (1/sqrt(C)),
// scattered into out[n, :] per the reference concat layout, + sc.
// Fragment layouts (ISA 7.12.2):
//   A (16x4): M = lane&15, K = vgpr_j + 2*(lane>=16)    -> contiguous float2
//   B (4x16): N = lane&15, K = vgpr_j + 2*(lane>=16)
//   C/D:      N = lane&15, M = vgpr_r + 8*(lane>=16)
// =====================================================================
__global__ __launch_bounds__(256)
void linear_wmma(const float* __restrict__ FB,
                 const float* __restrict__ Wlin0, const float* __restrict__ Wlin1,
                 const float* __restrict__ sc, float* __restrict__ out) {
    int wave = threadIdx.x >> 5;
    int lane = threadIdx.x & 31;
    int tile = blockIdx.x * 8 + wave;        // 4096 tiles total
    int wt = tile & 7;                       // w-tile (C/16)
    int nt = (tile >> 3) & 127;              // n-tile (N/16)
    int d  = tile >> 10;                     // 0..3

    int half = lane >> 4;
    int m16  = lane & 15;
    int nbase = nt * 16, wbase = wt * 16;

    const float* Fd = FB + (size_t)d * N_ * C_;
    const float* Wl = (d == 0) ? Wlin0 : Wlin1;

    v8f acc = {};
    for (int kb = 0; kb < C_ / 4; ++kb) {
        int k0 = kb * 4 + 2 * half;
        v2f a = *(const v2f*)(Fd + (size_t)(nbase + m16) * C_ + k0);
        const float* bp = Wl + (size_t)k0 * C_ + wbase + m16;
        v2f bv; bv.x = bp[0]; bv.y = bp[C_];
        acc = __builtin_amdgcn_wmma_f32_16x16x4_f32(
            /*neg_a=*/false, a, /*neg_b=*/false, bv,
            /*c_mod=*/(short)0, acc, /*reuse_a=*/false, /*reuse_b=*/false);
    }

    const float inv = 0.08838834764831845f;  // 1/sqrt(128)
    #pragma unroll
    for (int r = 0; r < 8; ++r) {
        int row = nbase + r + 8 * half;
        int col = wbase + m16;
        size_t off = (size_t)row * OUTC +
                     ((d == 0) ? (size_t)col : (size_t)(C_ + col * 3 + (d - 1)));
        out[off] = sc[off] + acc[r] * inv;
    }
}

// =====================================================================
// launcher
// =====================================================================
extern "C" void kernel_launch(void* const* d_in, const int* in_sizes, int n_in,
                              void* d_out, int out_size, void* d_ws, size_t ws_size,
                              hipStream_t stream) {
    // inputs in setup_inputs() dict order
    const float* node_feats = (const float*)d_in[0];
    const float* sc         = (const float*)d_in[1];
    const float* node_attrs = (const float*)d_in[2];
    const float* U3_0 = (const float*)d_in[3];
    const float* U2_0 = (const float*)d_in[4];
    const float* U1_0 = (const float*)d_in[5];
    const float* W3_0 = (const float*)d_in[6];
    const float* W2_0 = (const float*)d_in[7];
    const float* W1_0 = (const float*)d_in[8];
    const float* U3_1 = (const float*)d_in[9];
    const float* U2_1 = (const float*)d_in[10];
    const float* U1_1 = (const float*)d_in[11];
    const float* W3_1 = (const float*)d_in[12];
    const float* W2_1 = (const float*)d_in[13];
    const float* W1_1 = (const float*)d_in[14];
    const float* Wlin0 = (const float*)d_in[15];
    const float* Wlin1 = (const float*)d_in[16];
    float* out = (float*)d_out;

    // workspace carve-up (~21 MB)
    float* A3 = (float*)d_ws;
    float* A2 = A3 + A3_ELEMS;
    float* A1 = A2 + A2_ELEMS;
    float* FB = A1 + A1_ELEMS;
    int* counts = (int*)(FB + FB_ELEMS);
    int* lists  = counts + 16;

    zero_counts<<<1, 32, 0, stream>>>(counts);
    bucket_nodes<<<N_ / 256, 256, 0, stream>>>(node_attrs, counts, lists);
    build_A3<<<(int)(A3_ELEMS / 256), 256, 0, stream>>>(U3_0, W3_0, U3_1, W3_1, A3);
    build_A2<<<(int)(A2_ELEMS / 256), 256, 0, stream>>>(U2_0, W2_0, U2_1, W2_1, A2);
    build_A1<<<(int)(A1_ELEMS / 256), 256, 0, stream>>>(U1_0, W1_0, U1_1, W1_1, A1);
    sym_contract<<<E_ * 8 * 16, 256, 0, stream>>>(node_feats, counts, lists, A3, A2, A1, FB);
    linear_wmma<<<(DTOT * (N_ / 16) * (C_ / 16)) / 8, 256, 0, stream>>>(FB, Wlin0, Wlin1, sc, out);
}